// WeightedSegmentationLoss_5566277616018
// MI455X (gfx1250) — compile-verified
//
#include <hip/hip_runtime.h>
#include <stdint.h>

// ---------------------------------------------------------------------------
// WeightedSegmentationLoss for MI455X (gfx1250, wave32)
//   pred   : float[4,6,384,384]
//   target : int[4,384,384]
//   out    : float[1]  (mean weighted loss)
//
// Pass 1: per-column 1D distance scan (matches reference BIG=1e6 carry).
// Pass 2: per-row exact EDT lower envelope out of LDS + fused cross-entropy
//         + background Gaussian weighting + block reduction. LDS staging uses
//         GLOBAL_LOAD_ASYNC_TO_LDS_B128 (ASYNCcnt) overlapped with the six
//         strided pred-plane register loads.
// Pass 3: single-wave exact f32 reduction using V_WMMA_F32_16X16X4_F32
//         (D = A x ones + C gives exact f32 row sums on the matrix pipe).
// ---------------------------------------------------------------------------

#define NCLS   6
#define BGCLS  5
#define NB     4
#define NH     384
#define NW     384
#define HW     (NH * NW)
#define NPIX   (NB * NH * NW)
#define NROWS  (NB * NH)          // 1536
#define INV2S2 (1.0f / 200.0f)    // 1/(2*sigma^2), sigma=10
#define BIGD   1.0e6f

typedef __attribute__((ext_vector_type(2))) float v2f;
typedef __attribute__((ext_vector_type(8))) float v8f;
typedef int v4i __attribute__((vector_size(16)));   // matches builtin's pointee

#define AS_GLOBAL __attribute__((address_space(1)))
#define AS_LDS    __attribute__((address_space(3)))

#if defined(__has_builtin)
#  if __has_builtin(__builtin_amdgcn_global_load_async_to_lds_b128)
#    define HAVE_ASYNC_LDS 1
#  endif
#endif

// ---------------------------------------------------------------------------
// Pass 1: column-wise 1D distance (forward + backward scan), store squared.
// One thread per (b, w) column; accesses coalesced across w.
// ---------------------------------------------------------------------------
__global__ __launch_bounds__(256)
void col_dist_kernel(const int* __restrict__ tgt, float* __restrict__ g2) {
    int t = blockIdx.x * blockDim.x + threadIdx.x;
    if (t >= NB * NW) return;
    int b = t / NW, w = t % NW;
    const int* tb = tgt + (size_t)b * HW + w;
    float*     gb = g2  + (size_t)b * HW + w;

    float d = BIGD;                       // reference: carry init = 1e6
    for (int h = 0; h < NH; ++h) {
        bool f = (tb[h * NW] != BGCLS);
        d = f ? 0.0f : d + 1.0f;
        gb[h * NW] = d;                   // forward distances
    }
    d = BIGD;
    for (int h = NH - 1; h >= 0; --h) {
        bool f = (tb[h * NW] != BGCLS);
        d = f ? 0.0f : d + 1.0f;
        float m = fminf(gb[h * NW], d);   // min(fwd, bwd)
        gb[h * NW] = m * m;               // squared column distance
    }
}

// ---------------------------------------------------------------------------
// Pass 2: one block per image row (b, i). 384 threads, one pixel j each.
//   d2[j] = min_k ( g2[k] + (j-k)^2 )    -- exact EDT lower envelope from LDS
//   loss  = -log_softmax(pred)[target]
//   val   = (target==bg) ? loss * exp(-d2/(2s^2)) : loss
// Block-reduce -> partial[row].
// ---------------------------------------------------------------------------
__global__ __launch_bounds__(NW)
void row_loss_kernel(const float* __restrict__ pred,
                     const int*   __restrict__ tgt,
                     const float* __restrict__ g2,
                     float*       __restrict__ partial) {
    __shared__ __align__(16) float sg2[NW];
    __shared__ float red[512];

    int row = blockIdx.x;                 // 0..NROWS-1
    int b = row / NH, i = row % NH;
    int j = threadIdx.x;                  // 0..383

    const float* grow = g2 + (size_t)b * HW + (size_t)i * NW;

    // --- stage g2 row into LDS ---------------------------------------------
#ifdef HAVE_ASYNC_LDS
    // 96 lanes DMA 16 bytes each, global -> LDS, no VGPR round-trip.
    if (j < NW / 4) {
        __builtin_amdgcn_global_load_async_to_lds_b128(
            (AS_GLOBAL v4i*)(uintptr_t)(grow + 4 * j),
            (AS_LDS    v4i*)(uint32_t)(uintptr_t)(&sg2[4 * j]),
            /*offset=*/0, /*cpol=*/0);
    }
#else
    sg2[j] = grow[j];
#endif

    // --- overlap: issue the six strided pred loads + target while the LDS
    //     staging is in flight ----------------------------------------------
    const float* p = pred + ((size_t)b * NCLS * NH + i) * NW + j;
    float x[NCLS];
#pragma unroll
    for (int c = 0; c < NCLS; ++c) x[c] = p[(size_t)c * HW];
    int t = tgt[(size_t)b * HW + (size_t)i * NW + j];

    if (j < 512 - NW) red[NW + j] = 0.0f; // zero the reduction pad

#ifdef HAVE_ASYNC_LDS
#  if __has_builtin(__builtin_amdgcn_s_wait_asynccnt)
    __builtin_amdgcn_s_wait_asynccnt(0);
#  else
    asm volatile("s_wait_asynccnt 0x0" ::: "memory");
#  endif
#endif
    __syncthreads();

    // --- exact lower envelope (min-plus); LDS reads broadcast, conflict-free
    float dj = (float)j;
    float m = 3.0e38f;
#pragma unroll 8
    for (int k = 0; k < NW; ++k) {
        float diff = dj - (float)k;
        m = fminf(m, __builtin_fmaf(diff, diff, sg2[k]));
    }

    // --- cross entropy over 6 classes (values already in registers) --------
    float mx = -3.0e38f;
#pragma unroll
    for (int c = 0; c < NCLS; ++c) mx = fmaxf(mx, x[c]);
    float se = 0.0f;
#pragma unroll
    for (int c = 0; c < NCLS; ++c) se += __expf(x[c] - mx);
    float loss = -(x[t] - mx - __logf(se));

    float val = (t == BGCLS) ? loss * __expf(-m * INV2S2) : loss;

    // --- padded tree reduction (384 -> 1 via 512-wide array) ---------------
    red[j] = val;
    __syncthreads();
#pragma unroll
    for (int s = 256; s > 0; s >>= 1) {
        if (j < s) red[j] += red[j + s];
        __syncthreads();
    }
    if (j == 0) partial[row] = red[0];
}

// ---------------------------------------------------------------------------
// Pass 3: single wave32. Reduce 1536 partials with 24 accumulating
// V_WMMA_F32_16X16X4_F32 ops: A = 16x4 tile of partials (v2f/lane),
// B = ones => D[m,n] = rowsum(A_m) + C[m,n], exact f32 adds on matrix pipe.
// Sum of all D elements = 16 * total. Wave fold + scale + write mean.
// ---------------------------------------------------------------------------
__global__ __launch_bounds__(32)
void final_reduce_kernel(const float* __restrict__ partial,
                         float* __restrict__ out) {
    int lane = threadIdx.x;               // 0..31, all lanes active (EXEC==~0)
    v8f c = {};
    v2f bones;
    bones[0] = 1.0f;
    bones[1] = 1.0f;

    int mrow  = lane & 15;                // A-matrix row this lane holds
    int khalf = (lane >> 4) << 1;         // lanes 0-15: K=0,1 ; 16-31: K=2,3

#pragma unroll
    for (int it = 0; it < NROWS / 64; ++it) {   // 24 iterations, 64 values each
        const float* base = partial + it * 64;  // A[m][k] = base[m*4+k]
        v2f a;
        a[0] = base[mrow * 4 + khalf + 0];
        a[1] = base[mrow * 4 + khalf + 1];
        c = __builtin_amdgcn_wmma_f32_16x16x4_f32(
                /*neg_a=*/false, a, /*neg_b=*/false, bones,
                /*c_mod=*/(short)0, c, /*reuse_a=*/false, /*reuse_b=*/false);
    }

    float s = c[0] + c[1] + c[2] + c[3] + c[4] + c[5] + c[6] + c[7];
#pragma unroll
    for (int off = 16; off > 0; off >>= 1)
        s += __shfl_xor(s, off, 32);

    if (lane == 0)
        out[0] = (s / 16.0f) / (float)NPIX;
}

// ---------------------------------------------------------------------------
extern "C" void kernel_launch(void* const* d_in, const int* in_sizes, int n_in,
                              void* d_out, int out_size, void* d_ws, size_t ws_size,
                              hipStream_t stream) {
    const float* pred = (const float*)d_in[0];
    const int*   tgt  = (const int*)d_in[1];
    float* g2      = (float*)d_ws;        // NPIX floats (2.36 MB)
    float* partial = g2 + NPIX;           // NROWS floats
    float* out     = (float*)d_out;

    col_dist_kernel<<<(NB * NW + 255) / 256, 256, 0, stream>>>(tgt, g2);
    row_loss_kernel<<<NROWS, NW, 0, stream>>>(pred, tgt, g2, partial);
    final_reduce_kernel<<<1, 32, 0, stream>>>(partial, out);
}